// AvgpoolLayerWithoutselfloop_19146964206243
// MI455X (gfx1250) — compile-verified
//
#include <hip/hip_runtime.h>
#include <hip/hip_bf16.h>

// Scatter-mean (graph neighbor average pool) for MI455X / gfx1250.
//
// x:          [50000, 64] f32   (12.8 MB -> fully L2-resident, ~16x reuse)
// edge_index: [2, 800000] i32   (6.4 MB, streamed once -> non-temporal loads)
// out[c,:] = mean over edges e with col[e]==c of x[row[e],:]  (empty -> sum/1)
//
// Roofline: ~55 MFLOP vs ~420 MB of L2 traffic -> purely L2-atomic bound.
// Strategy: L2-resident gather + fire-and-forget global_atomic_add_f32
// (non-returning, STOREcnt path), 16 lanes per edge (one b128 load + 4 b32
// atomics per lane), gfx1250 global_prefetch_b8 on a future gather row,
// NT temporal hints to keep the reused x table hot in the WGP cache.

#define EDGE_PREFETCH_DIST 1024

typedef float vfloat4 __attribute__((ext_vector_type(4)));

__global__ __launch_bounds__(256)
void gnn_avgpool_zero_kernel(vfloat4* __restrict__ out4,
                             float*   __restrict__ cnt,
                             int n_out4, int n_nodes) {
    int tid = blockIdx.x * blockDim.x + threadIdx.x;
    if (tid < n_out4)  out4[tid] = (vfloat4){0.f, 0.f, 0.f, 0.f};
    if (tid < n_nodes) cnt[tid]  = 0.f;
}

__global__ __launch_bounds__(256)
void gnn_avgpool_scatter_kernel(const vfloat4* __restrict__ x4,      // x as float4: [n_nodes*16]
                                const int*     __restrict__ row_idx, // edge_index[0,:]
                                const int*     __restrict__ col_idx, // edge_index[1,:]
                                float*         __restrict__ out,     // sums, [n_nodes*64]
                                float*         __restrict__ cnt,     // [n_nodes]
                                int n_edges) {
    int tid  = blockIdx.x * 256 + threadIdx.x;
    int e    = tid >> 4;        // 16 lanes cooperate on one edge
    int lane = tid & 15;        // lane handles features [4*lane, 4*lane+4)
    if (e >= n_edges) return;

    // Edge stream is read-once: non-temporal so it never evicts x rows from WGP$.
    // All 16 lanes of a group hit the same cacheline -> coalesces to 1 request.
    int row = __builtin_nontemporal_load(&row_idx[e]);
    int col = __builtin_nontemporal_load(&col_idx[e]);

    if (lane == 0) {
        // Prefetch the gather row of a future edge (gfx1250 global_prefetch_b8)
        // to hide the data-dependent x[row] latency behind the atomic stream.
        int pe = e + EDGE_PREFETCH_DIST;
        if (pe < n_edges) {
            int frow = __builtin_nontemporal_load(&row_idx[pe]);
            __builtin_prefetch(&x4[(size_t)frow * 16], 0, 3);
        }
        // degree count, fused: one atomic per edge
        atomicAdd(&cnt[col], 1.0f);
    }

    // b128 gather of this lane's 4 features (temporal: x is the hot, reused table)
    vfloat4 v = x4[(size_t)row * 16 + lane];

    // fire-and-forget global_atomic_add_f32 (no return -> STOREcnt, no RAW stall)
    float* dst = out + (size_t)col * 64 + lane * 4;
    atomicAdd(dst + 0, v.x);
    atomicAdd(dst + 1, v.y);
    atomicAdd(dst + 2, v.z);
    atomicAdd(dst + 3, v.w);
}

__global__ __launch_bounds__(256)
void gnn_avgpool_divide_kernel(vfloat4* __restrict__ out4,
                               const float* __restrict__ cnt,
                               int n_nodes) {
    int tid = blockIdx.x * blockDim.x + threadIdx.x;
    if (tid >= n_nodes * 16) return;
    int node = tid >> 4;
    float c   = fmaxf(cnt[node], 1.0f);   // PyG scatter-mean clamps empty segments
    float inv = 1.0f / c;
    // sums are last-use, output is write-once: keep the epilogue out of the caches
    vfloat4 v = __builtin_nontemporal_load(&out4[tid]);
    v *= inv;
    __builtin_nontemporal_store(v, &out4[tid]);
}

extern "C" void kernel_launch(void* const* d_in, const int* in_sizes, int n_in,
                              void* d_out, int out_size, void* d_ws, size_t ws_size,
                              hipStream_t stream) {
    const float* x  = (const float*)d_in[0];   // [n_nodes, 64] f32
    const int*   ei = (const int*)d_in[1];     // [2, n_edges] i32, row-major

    const int n_nodes = in_sizes[0] / 64;
    const int n_edges = in_sizes[1] / 2;

    const int* row_idx = ei;                   // ei[0, :]
    const int* col_idx = ei + n_edges;         // ei[1, :]

    float*   out  = (float*)d_out;             // [n_nodes, 64] sums -> means (in place)
    vfloat4* out4 = (vfloat4*)d_out;
    float*   cnt  = (float*)d_ws;              // [n_nodes] degree counts (200 KB scratch)

    const int n_out4 = n_nodes * 16;           // 64 feats = 16 float4 per node

    // 1) zero sums + counts
    {
        int threads = 256;
        int blocks  = (n_out4 + threads - 1) / threads;   // n_out4 > n_nodes
        gnn_avgpool_zero_kernel<<<blocks, threads, 0, stream>>>(out4, cnt, n_out4, n_nodes);
    }

    // 2) gather + scatter-add (16 lanes per edge)
    {
        int threads = 256;
        long long total = (long long)n_edges * 16;
        int blocks = (int)((total + threads - 1) / threads);
        gnn_avgpool_scatter_kernel<<<blocks, threads, 0, stream>>>(
            (const vfloat4*)x, row_idx, col_idx, out, cnt, n_edges);
    }

    // 3) divide by clamped counts
    {
        int threads = 256;
        int blocks  = (n_out4 + threads - 1) / threads;
        gnn_avgpool_divide_kernel<<<blocks, threads, 0, stream>>>(out4, cnt, n_nodes);
    }
}